// WindAdaptivePooling_30305289241063
// MI455X (gfx1250) — compile-verified
//
#include <hip/hip_runtime.h>
#include <hip/hip_bf16.h>

typedef float v2f __attribute__((ext_vector_type(2)));
typedef float v8f __attribute__((ext_vector_type(8)));

// ---- CDNA5 async global->LDS copy (B128: 16 B per lane, 512 B per wave) ----
// GV mode: global_load_async_to_lds_b128 vdst(lds byte addr), vaddr(64b), off
// Flat pointers to LDS carry the LDS byte offset in their low 32 bits (ISA 10.2),
// so a plain truncation produces the VDST operand.
__device__ static __forceinline__ void async_ld_b128(const float* g, const float* l) {
  unsigned loff = (unsigned)(size_t)l;
  asm volatile("global_load_async_to_lds_b128 %0, %1, off"
               :: "v"(loff), "v"(g) : "memory");
}

template <int N>
__device__ static __forceinline__ void wait_asynccnt() {
#if __has_builtin(__builtin_amdgcn_s_wait_asynccnt)
  __builtin_amdgcn_s_wait_asynccnt(N);
#else
  asm volatile("s_wait_asynccnt %0" :: "i"(N) : "memory");
#endif
}

template <int N>
__device__ static __forceinline__ void wait_dscnt() {
#if __has_builtin(__builtin_amdgcn_s_wait_dscnt)
  __builtin_amdgcn_s_wait_dscnt(N);
#else
  asm volatile("s_wait_dscnt %0" :: "i"(N) : "memory");
#endif
}

// D = A(16x4 f32) * B(4x16 f32) + C  — B = ones => every D column is the row sum.
__device__ static __forceinline__ v8f wmma_rowsum(v2f a, v2f ones, v8f c) {
  return __builtin_amdgcn_wmma_f32_16x16x4_f32(
      /*neg_a=*/false, a, /*neg_b=*/false, ones,
      /*c_mod=*/(short)0, c, /*reuse_a=*/false, /*reuse_b=*/false);
}

// Sizes fixed by the reference: B=64, C=1280, H*W=1024, NUM_WINDS=8.
#define HW        1024
#define CCH       1280
#define ROW_DW    132            // 128 data dwords + 4 pad (bank-conflict-free A loads)
#define BUF_DW    (16 * ROW_DW)  // one 16-row x 128-col chunk = 2112 dwords
#define WAVE_DW   (2 * BUF_DW)   // double buffer per wave

__global__ __launch_bounds__(256)
void wind_pool_kernel(const float* __restrict__ x,
                      const int*   __restrict__ widx,
                      const float* __restrict__ ww,
                      float*       __restrict__ out) {
  __shared__ __align__(16) float lds[8 * WAVE_DW];  // 8 waves * 16896 B = 135168 B

  const int tid  = threadIdx.x;
  const int lane = tid & 31;
  const int w    = tid >> 5;
  const int hi   = lane >> 4;   // 0: K cols +0..1, 1: K cols +2..3 (A layout)
  const int m    = lane & 15;   // row within tile
  const int tile = (blockIdx.x * 8 + w) * 16;  // 16 rows per wave, 640 blocks total

  float* l0 = &lds[w * WAVE_DW];
  float* l1 = l0 + BUF_DW;
  const float* gbase = x + (size_t)tile * HW + lane * 4;  // lane*16 B within a row-chunk

  // Prologue: stream chunk 0 into buf0 (16 async b128 = 8 KB tile-chunk).
#pragma unroll
  for (int i = 0; i < 16; ++i)
    async_ld_b128(gbase + i * HW, l0 + i * ROW_DW + lane * 4);

  v8f c0 = {0.f,0.f,0.f,0.f,0.f,0.f,0.f,0.f};
  v8f c1 = {0.f,0.f,0.f,0.f,0.f,0.f,0.f,0.f};
  const v2f ones = {1.0f, 1.0f};

  for (int ck = 0; ck < 8; ++ck) {
    float* cur = (ck & 1) ? l1 : l0;
    float* nxt = (ck & 1) ? l0 : l1;

    if (ck < 7) {
      wait_dscnt<0>();  // previous-iter LDS reads finished before overwriting nxt
#pragma unroll
      for (int i = 0; i < 16; ++i)
        async_ld_b128(gbase + i * HW + (ck + 1) * 128, nxt + i * ROW_DW + lane * 4);
      wait_asynccnt<16>();  // oldest 16 (current chunk) landed; next 16 still in flight
    } else {
      wait_asynccnt<0>();
    }

    // A-operand: lane m reads row m, cols k + 2*hi (ds_load_b64, conflict-free).
    const float* ap = cur + m * ROW_DW + hi * 2;
#pragma unroll
    for (int kk = 0; kk < 32; kk += 2) {
      v2f a0 = *(const v2f*)(ap + kk * 4);
      v2f a1 = *(const v2f*)(ap + kk * 4 + 4);
      c0 = wmma_rowsum(a0, ones, c0);
      c1 = wmma_rowsum(a1, ones, c1);
    }
  }

  v8f cc = c0 + c1;  // merge K-partition accumulators

  // D layout: VGPR j -> row j (lanes 0-15) / row 8+j (lanes 16-31); all N equal.
  if (m == 0) {  // lanes 0 and 16 each own 8 rows
    const int b = tile / CCH;  // tile never crosses batch boundary (1280 % 16 == 0)
    float mx = ww[0];
#pragma unroll
    for (int i = 1; i < 8; ++i) mx = fmaxf(mx, ww[i]);
    float s = 0.f;
#pragma unroll
    for (int i = 0; i < 8; ++i) s += __expf(ww[i] - mx);
    const float scale = __expf(ww[widx[b]] - mx) / s;
    const float f = scale * (1.0f / (float)HW);

    const int r = tile + hi * 8;
    float4 o0 = {cc[0]*f, cc[1]*f, cc[2]*f, cc[3]*f};
    float4 o1 = {cc[4]*f, cc[5]*f, cc[6]*f, cc[7]*f};
    *(float4*)(out + r)     = o0;
    *(float4*)(out + r + 4) = o1;
  }
}

extern "C" void kernel_launch(void* const* d_in, const int* in_sizes, int n_in,
                              void* d_out, int out_size, void* d_ws, size_t ws_size,
                              hipStream_t stream) {
  const float* x    = (const float*)d_in[0];
  const int*   widx = (const int*)d_in[1];
  const float* ww   = (const float*)d_in[2];
  float*       out  = (float*)d_out;
  (void)in_sizes; (void)n_in; (void)out_size; (void)d_ws; (void)ws_size;

  // 81920 rows / (8 waves * 16 rows) = 640 blocks of 256 threads.
  wind_pool_kernel<<<dim3(640), dim3(256), 0, stream>>>(x, widx, ww, out);
}